// GCN_60335700574378
// MI455X (gfx1250) — compile-verified
//
#include <hip/hip_runtime.h>
#include <math.h>

// ---------------------------------------------------------------------------
// GCN forward (2x GCNConv + PReLU + log_softmax) for MI455X / gfx1250.
// GEMMs use V_WMMA_F32_16X16X4_F32 (exact fp32, matrix pipe). Aggregation is
// L2-resident scatter-add with f32 global atomics.
// ---------------------------------------------------------------------------

typedef __attribute__((ext_vector_type(2))) float v2f;
typedef __attribute__((ext_vector_type(8))) float v8f;

#define GCN_NODES 50000
#define GCN_EDGES 1200000
#define GCN_IN    256
#define GCN_HID   64

// ---------------- degree / normalization ----------------

__global__ void k_init_deg(float* __restrict__ deg, int n) {
    int i = blockIdx.x * blockDim.x + threadIdx.x;
    if (i < n) deg[i] = 1.0f;   // self-loop contributes 1 to in-degree
}

__global__ void k_accum_deg(const int* __restrict__ dst, float* __restrict__ deg, int e) {
    int i = blockIdx.x * blockDim.x + threadIdx.x;
    if (i < e) atomicAdd(&deg[dst[i]], 1.0f);
}

__global__ void k_deg_to_dinv(float* __restrict__ deg, int n) {
    int i = blockIdx.x * blockDim.x + threadIdx.x;
    if (i < n) {
        float d = deg[i];
        deg[i] = (d > 0.0f) ? rsqrtf(d) : 0.0f;
    }
}

// ---------------- fp32 WMMA GEMM: C[M x 64] = A[M x K] @ B[K x 64] ----------

template <int K>
__global__ __launch_bounds__(256) void k_gemm_wmma_f32(const float* __restrict__ A,
                                                       const float* __restrict__ B,
                                                       float* __restrict__ C, int M) {
    const int wave = threadIdx.x >> 5;          // 8 waves / block
    const int lane = threadIdx.x & 31;
    const int tile = blockIdx.x * 8 + wave;     // one 16-row tile per wave
    if (tile * 16 >= M) return;                 // wave-uniform: EXEC stays all-1s

    const int row0 = tile * 16;
    const int m    = lane & 15;                 // M (A) / N (B) index for this lane
    const int kb   = (lane >> 4) << 1;          // K sub-offset: 0 or 2

    v8f acc0 = {}, acc1 = {}, acc2 = {}, acc3 = {};
    const float* __restrict__ arow = A + (size_t)(row0 + m) * K;

    for (int k0 = 0; k0 < K; k0 += 4) {
        v2f af;
        af.x = arow[k0 + kb];
        af.y = arow[k0 + kb + 1];
        const float* __restrict__ brow0 = B + (size_t)(k0 + kb) * 64;
        const float* __restrict__ brow1 = B + (size_t)(k0 + kb + 1) * 64;

        v2f bf;
        bf.x = brow0[m];      bf.y = brow1[m];
        acc0 = __builtin_amdgcn_wmma_f32_16x16x4_f32(false, af, false, bf, (short)0, acc0, false, false);
        bf.x = brow0[16 + m]; bf.y = brow1[16 + m];
        acc1 = __builtin_amdgcn_wmma_f32_16x16x4_f32(false, af, false, bf, (short)0, acc1, false, false);
        bf.x = brow0[32 + m]; bf.y = brow1[32 + m];
        acc2 = __builtin_amdgcn_wmma_f32_16x16x4_f32(false, af, false, bf, (short)0, acc2, false, false);
        bf.x = brow0[48 + m]; bf.y = brow1[48 + m];
        acc3 = __builtin_amdgcn_wmma_f32_16x16x4_f32(false, af, false, bf, (short)0, acc3, false, false);
    }

    // D layout: VGPR j holds row M = j + (lane/16)*8, col N = lane%16
    float* __restrict__ crow = C + (size_t)row0 * 64;
    const int mo = (lane >> 4) * 8;
#pragma unroll
    for (int j = 0; j < 8; ++j) {
        crow[(size_t)(j + mo) * 64 + m]      = acc0[j];
        crow[(size_t)(j + mo) * 64 + 16 + m] = acc1[j];
        crow[(size_t)(j + mo) * 64 + 32 + m] = acc2[j];
        crow[(size_t)(j + mo) * 64 + 48 + m] = acc3[j];
    }
}

// ---------------- aggregation: agg[dst] += H[src] * dinv[src]*dinv[dst] -----

// Self-loop seed: agg[i, :] = H[i, :] * dinv[i]^2
__global__ void k_selfloop_init(const float* __restrict__ H, const float* __restrict__ dinv,
                                float* __restrict__ agg, int total) {
    int gid = blockIdx.x * blockDim.x + threadIdx.x;
    if (gid < total) {
        int node = gid >> 6;
        float di = dinv[node];
        agg[gid] = H[gid] * di * di;
    }
}

// One wave32 per edge, float2 per lane (64 dims), two f32 atomics per lane.
__global__ __launch_bounds__(256) void k_edge_agg(const int* __restrict__ src,
                                                  const int* __restrict__ dst,
                                                  const float* __restrict__ dinv,
                                                  const float* __restrict__ H,
                                                  float* __restrict__ agg, int e) {
    int edge = blockIdx.x * 8 + (threadIdx.x >> 5);
    if (edge >= e) return;
    int lane = threadIdx.x & 31;
    int s = src[edge];
    int t = dst[edge];
    float w = dinv[s] * dinv[t];
    const v2f hv = *(const v2f*)(H + (size_t)s * 64 + lane * 2);
    float* p = agg + (size_t)t * 64 + lane * 2;
    atomicAdd(p,     hv.x * w);
    atomicAdd(p + 1, hv.y * w);
}

// ---------------- epilogues ----------------

__global__ void k_bias_prelu(const float* __restrict__ agg, const float* __restrict__ bias,
                             const float* __restrict__ a, float* __restrict__ out, int total) {
    int gid = blockIdx.x * blockDim.x + threadIdx.x;
    if (gid < total) {
        float alpha = a[0];
        float v = agg[gid] + bias[gid & 63];
        out[gid] = (v >= 0.0f) ? v : alpha * v;
    }
}

// Fused bias + PReLU + log_softmax over 64 dims. One wave32 per node.
__global__ __launch_bounds__(256) void k_prelu_logsoftmax(const float* __restrict__ agg,
                                                          const float* __restrict__ bias,
                                                          const float* __restrict__ a,
                                                          float* __restrict__ out, int n) {
    int node = blockIdx.x * 8 + (threadIdx.x >> 5);
    if (node >= n) return;
    int lane = threadIdx.x & 31;
    float alpha = a[0];

    v2f v = *(const v2f*)(agg + (size_t)node * 64 + lane * 2);
    v.x += bias[2 * lane];
    v.y += bias[2 * lane + 1];
    v.x = (v.x >= 0.0f) ? v.x : alpha * v.x;
    v.y = (v.y >= 0.0f) ? v.y : alpha * v.y;

    float mx = fmaxf(v.x, v.y);
#pragma unroll
    for (int off = 16; off > 0; off >>= 1)
        mx = fmaxf(mx, __shfl_xor(mx, off, 32));

    float s = expf(v.x - mx) + expf(v.y - mx);
#pragma unroll
    for (int off = 16; off > 0; off >>= 1)
        s += __shfl_xor(s, off, 32);

    float lse = mx + logf(s);
    float* o = out + (size_t)node * 64 + lane * 2;
    o[0] = v.x - lse;
    o[1] = v.y - lse;
}

// ---------------- launch ----------------

extern "C" void kernel_launch(void* const* d_in, const int* in_sizes, int n_in,
                              void* d_out, int out_size, void* d_ws, size_t ws_size,
                              hipStream_t stream) {
    (void)in_sizes; (void)n_in; (void)out_size; (void)ws_size;

    const float* x    = (const float*)d_in[0];
    const int*   eidx = (const int*)d_in[1];
    const int*   src  = eidx;                 // edge_index[0]
    const int*   dst  = eidx + GCN_EDGES;     // edge_index[1]
    const float* W1   = (const float*)d_in[2];
    const float* b1   = (const float*)d_in[3];
    const float* W2   = (const float*)d_in[4];
    const float* b2   = (const float*)d_in[5];
    const float* a    = (const float*)d_in[6];
    float* out        = (float*)d_out;

    // workspace layout (floats)
    float* ws   = (float*)d_ws;
    float* dinv = ws;                                   // [N]      (deg, then dinv)
    float* Hbuf = dinv + GCN_NODES;                     // [N x 64] GEMM output
    float* Agg  = Hbuf + (size_t)GCN_NODES * 64;        // [N x 64] aggregation
    float* Act  = Agg  + (size_t)GCN_NODES * 64;        // [N x 64] activated hidden

    const int feat_total = GCN_NODES * 64;
    const int tiles      = GCN_NODES / 16;              // 3125, exact
    const dim3 gemm_grid((tiles + 7) / 8);
    const dim3 edge_grid((GCN_EDGES + 7) / 8);
    const dim3 node_grid((GCN_NODES + 7) / 8);

    // normalization: deg -> dinv
    k_init_deg<<<(GCN_NODES + 255) / 256, 256, 0, stream>>>(dinv, GCN_NODES);
    k_accum_deg<<<(GCN_EDGES + 255) / 256, 256, 0, stream>>>(dst, dinv, GCN_EDGES);
    k_deg_to_dinv<<<(GCN_NODES + 255) / 256, 256, 0, stream>>>(dinv, GCN_NODES);

    // layer 1: H = x @ W1 ; aggregate ; bias + PReLU
    k_gemm_wmma_f32<GCN_IN><<<gemm_grid, 256, 0, stream>>>(x, W1, Hbuf, GCN_NODES);
    k_selfloop_init<<<(feat_total + 255) / 256, 256, 0, stream>>>(Hbuf, dinv, Agg, feat_total);
    k_edge_agg<<<edge_grid, 256, 0, stream>>>(src, dst, dinv, Hbuf, Agg, GCN_EDGES);
    k_bias_prelu<<<(feat_total + 255) / 256, 256, 0, stream>>>(Agg, b1, a, Act, feat_total);

    // layer 2: H = Act @ W2 ; aggregate ; bias + PReLU + log_softmax
    k_gemm_wmma_f32<GCN_HID><<<gemm_grid, 256, 0, stream>>>(Act, W2, Hbuf, GCN_NODES);
    k_selfloop_init<<<(feat_total + 255) / 256, 256, 0, stream>>>(Hbuf, dinv, Agg, feat_total);
    k_edge_agg<<<edge_grid, 256, 0, stream>>>(src, dst, dinv, Hbuf, Agg, GCN_EDGES);
    k_prelu_logsoftmax<<<node_grid, 256, 0, stream>>>(Agg, b2, a, out, GCN_NODES);
}